// ChamferPccRateDistortionLoss_25280177504595
// MI455X (gfx1250) — compile-verified
//
#include <hip/hip_runtime.h>
#include <math.h>

typedef __attribute__((ext_vector_type(2))) float v2f;
typedef __attribute__((ext_vector_type(8))) float v8f;

#define WAVES_PER_BLOCK 8
#define LN2F 0.69314718055994530942f

// ---------------------------------------------------------------------------
// ws layout (floats): ws[0] = chamfer row-min sum (both dirs), ws[1] = bpp sum
// ---------------------------------------------------------------------------

__global__ void init_ws_kernel(float* ws) {
    if (threadIdx.x < 2) ws[threadIdx.x] = 0.0f;
}

// Sum of -log(likelihoods); grid-stride, wave-reduce, one atomic per wave.
__global__ void bpp_sum_kernel(const float* __restrict__ lik, float* acc, int n) {
    int idx    = blockIdx.x * blockDim.x + threadIdx.x;
    int stride = gridDim.x * blockDim.x;
    float s = 0.0f;
    for (int i = idx; i < n; i += stride)
        s -= __logf(lik[i]);
    s += __shfl_xor(s, 16, 32);
    s += __shfl_xor(s,  8, 32);
    s += __shfl_xor(s,  4, 32);
    s += __shfl_xor(s,  2, 32);
    s += __shfl_xor(s,  1, 32);
    if ((threadIdx.x & 31) == 0)
        atomicAdd(acc, s);
}

// One-direction Chamfer: for each row p of X, min over q of ||x_p - y_q||^2,
// summed over all (b, p) into *acc. Cross term via V_WMMA_F32_16X16X4_F32.
__global__ __launch_bounds__(WAVES_PER_BLOCK * 32)
void chamfer_min_kernel(const float* __restrict__ X,
                        const float* __restrict__ Y,
                        float* acc, int P, int numTiles) {
    const int lane = threadIdx.x & 31;
    const int wave = threadIdx.x >> 5;
    const int tilesPerBatch = P >> 4;

    const int t = blockIdx.x * WAVES_PER_BLOCK + wave;   // wave-uniform
    if (t >= numTiles) return;                           // uniform branch: EXEC stays full

    const int b  = t / tilesPerBatch;
    const int m0 = (t - b * tilesPerBatch) << 4;         // first of 16 X rows
    const float* xb = X + (size_t)b * P * 3;
    const float* yb = Y + (size_t)b * P * 3;

    const bool hi  = lane >= 16;
    const int  l15 = lane & 15;

    // --- A operand: 16x4 f32 tile of X (K = {x,y,z,0}).
    // ISA layout: lanes 0-15 hold K=0,1 of row M=lane; lanes 16-31 hold K=2,3.
    {
    }
    const float* xp = xb + (size_t)(m0 + l15) * 3;
    const float xa0 = xp[0], xa1 = xp[1], xa2 = xp[2];
    v2f a;
    a.x = hi ? xa2 : xa0;
    a.y = hi ? 0.0f : xa1;

    // --- ||x||^2 for the 8 rows this lane's D registers cover:
    // D vgpr r, lane n: row = r + (n>=16 ? 8 : 0), col = n & 15.
    const int roff = hi ? 8 : 0;
    float x2r[8];
#pragma unroll
    for (int r = 0; r < 8; ++r) {
        const float* p = xb + (size_t)(m0 + roff + r) * 3;
        x2r[r] = p[0] * p[0] + p[1] * p[1] + p[2] * p[2];
    }

    float minv[8];
#pragma unroll
    for (int r = 0; r < 8; ++r) minv[r] = 3.402823466e38f;

    for (int q0 = 0; q0 < P; q0 += 16) {
        // B operand: 4x16 f32 tile of Y, column = l15; same split as A.
        const float* yp = yb + (size_t)(q0 + l15) * 3;
        const float y0 = yp[0], y1 = yp[1], y2c = yp[2];
        const float y2 = y0 * y0 + y1 * y1 + y2c * y2c;   // matches this lane's column
        v2f bv;
        bv.x = hi ? y2c : y0;
        bv.y = hi ? 0.0f : y1;

        v8f c = {};
        // D = A(16x4) * B(4x16): cross terms x_p . y_q for the 16x16 tile
        c = __builtin_amdgcn_wmma_f32_16x16x4_f32(
                /*neg_a=*/false, a, /*neg_b=*/false, bv,
                /*c_mod=*/(short)0, c, /*reuse_a=*/false, /*reuse_b=*/false);

#pragma unroll
        for (int r = 0; r < 8; ++r) {
            float d = x2r[r] + y2 - 2.0f * c[r];
            minv[r] = fminf(minv[r], d);
        }
    }

    // Min across the 16 lanes sharing each row (xor masks stay in half-wave).
#pragma unroll
    for (int r = 0; r < 8; ++r) {
        float m = minv[r];
        m = fminf(m, __shfl_xor(m, 1, 32));
        m = fminf(m, __shfl_xor(m, 2, 32));
        m = fminf(m, __shfl_xor(m, 4, 32));
        m = fminf(m, __shfl_xor(m, 8, 32));
        minv[r] = m;
    }
    if (l15 == 0) {   // lane 0 -> rows m0..m0+7, lane 16 -> rows m0+8..m0+15
        float s = 0.0f;
#pragma unroll
        for (int r = 0; r < 8; ++r) s += minv[r];
        atomicAdd(acc, s);
    }
}

__global__ void finalize_kernel(const float* __restrict__ ws, float* out, float invBP) {
    if (threadIdx.x == 0)
        out[0] = ws[1] * (invBP / LN2F) + ws[0] * invBP;
}

extern "C" void kernel_launch(void* const* d_in, const int* in_sizes, int n_in,
                              void* d_out, int out_size, void* d_ws, size_t ws_size,
                              hipStream_t stream) {
    const float* x_hat = (const float*)d_in[0];
    const float* pos   = (const float*)d_in[1];
    const float* lik   = (const float*)d_in[2];
    float* out = (float*)d_out;
    float* ws  = (float*)d_ws;

    const int nLik = in_sizes[2];              // B*256*512
    const int B    = nLik / (256 * 512);       // 8
    const int P    = in_sizes[0] / (3 * B);    // 4096

    init_ws_kernel<<<1, 32, 0, stream>>>(ws);

    {
        int threads = 256;
        int blocks  = 512;
        bpp_sum_kernel<<<blocks, threads, 0, stream>>>(lik, ws + 1, nLik);
    }

    const int numTiles = B * (P / 16);                                   // 2048
    const int blocks   = (numTiles + WAVES_PER_BLOCK - 1) / WAVES_PER_BLOCK;
    chamfer_min_kernel<<<blocks, WAVES_PER_BLOCK * 32, 0, stream>>>(
        x_hat, pos, ws, P, numTiles);   // cham_x: x -> y
    chamfer_min_kernel<<<blocks, WAVES_PER_BLOCK * 32, 0, stream>>>(
        pos, x_hat, ws, P, numTiles);   // cham_y: y -> x

    const float invBP = 1.0f / (float)(B * P);
    finalize_kernel<<<1, 32, 0, stream>>>(ws, out, invBP);
}